// GaussianBlur_33526514712581
// MI455X (gfx1250) — compile-verified
//
#include <hip/hip_runtime.h>

// -------- types for WMMA fragments (wave32, gfx1250) --------
typedef float v2f __attribute__((ext_vector_type(2)));
typedef float v8f __attribute__((ext_vector_type(8)));

// 1D Gaussian, ksize=7 sigma=2, normalized
__device__ constexpr float GK[7] = {
    0.0701593f, 0.1310749f, 0.1907128f, 0.2161058f,
    0.1907128f, 0.1310749f, 0.0701593f
};

// Problem constants: (N=2,C=2) -> 4 independent volumes of 160^3 f32
#define DIM     160
#define SLICES  (4 * DIM)           // (n*c) * z slices = 640
#define TILES_X (DIM / 16)          // 10
#define TILES_Y (DIM / 16)          // 10
#define TILES_PER_SLICE (TILES_X * TILES_Y)  // 100

// ============================================================================
// Pass 1: blur along W (contiguous axis) using V_WMMA_F32_16X16X4_F32.
//   Per 16x16 output tile: Out = In(16x24) x W(24x16), W banded with GK.
//   K=24 split into 6 chained WMMAs of K=4. One wave per tile, 4 waves/block.
// A-fragment layout (16x4 f32): lanes 0-15 hold K={0,1} in v[0],v[1] (M=lane),
//   lanes 16-31 hold K={2,3}. B mirrored (N=lane&15). C/D: VGPR r -> M=r+8*(lane>>4).
// ============================================================================
__global__ __launch_bounds__(128) void gb3d_blurx_wmma(
    const float* __restrict__ in, float* __restrict__ out) {
  const int wave  = threadIdx.x >> 5;
  const int lane  = threadIdx.x & 31;
  const int tile  = blockIdx.x * 4 + wave;          // 0 .. 63999, exact cover

  const int slice = tile / TILES_PER_SLICE;         // which (img,z) plane
  const int trem  = tile - slice * TILES_PER_SLICE;
  const int ty    = trem / TILES_X;
  const int tx    = trem - ty * TILES_X;
  const int x0    = tx * 16;
  const int y0    = ty * 16;

  const size_t base = (size_t)slice * (DIM * DIM);
  const int row = lane & 15;        // M for A-frag, N for B-frag, N for C/D
  const int h   = lane >> 4;        // half-wave selector

  const float* __restrict__ rowp = in + base + (size_t)(y0 + row) * DIM;

  v2f a[6], b[6];
#pragma unroll
  for (int c = 0; c < 6; ++c) {
#pragma unroll
    for (int e = 0; e < 2; ++e) {
      const int k  = 4 * c + 2 * h + e;       // 0..23 (K index)
      const int xi = x0 - 3 + k;              // global x with SAME padding
      const int xc = min(max(xi, 0), DIM - 1);
      // input streamed once -> non-temporal, keep L2 for live intermediates
      float av = __builtin_nontemporal_load(&rowp[xc]);
      a[c][e] = (xi == xc) ? av : 0.0f;       // zero-pad, no EXEC divergence
      const int d = k - row;                  // band of coefficient matrix
      b[c][e] = (d >= 0 && d <= 6) ? GK[d] : 0.0f;
    }
  }

  v8f acc = {0.f, 0.f, 0.f, 0.f, 0.f, 0.f, 0.f, 0.f};
#pragma unroll
  for (int c = 0; c < 6; ++c) {
    // (neg_a, A, neg_b, B, c_mod, C, reuse_a, reuse_b)
    acc = __builtin_amdgcn_wmma_f32_16x16x4_f32(
        false, a[c], false, b[c], (short)0, acc, false, false);
  }

  float* __restrict__ op = out + base + (size_t)(y0 + 8 * h) * DIM + x0 + row;
#pragma unroll
  for (int r = 0; r < 8; ++r)
    op[(size_t)r * DIM] = acc[r];             // lanes 0-15 / 16-31: two 64B rows
}

// ============================================================================
// Passes 2/3: 7-tap blur along y (stride 160 floats) or z (stride 25600),
// float4-vectorized along x -> fully coalesced 16B/lane streams.
// STRIDEF is a template constant so /,% strength-reduce to mul-hi.
// ============================================================================
template <int STRIDEF>
__global__ __launch_bounds__(256) void gb3d_blur_axis_f4(
    const float4* __restrict__ in, float4* __restrict__ out) {
  const long tid  = (long)blockIdx.x * 256 + threadIdx.x;  // float4 index
  const long flat = tid * 4;                                // float index
  const int  pos  = (int)((flat / STRIDEF) % DIM);          // coord on blur axis
  constexpr int S4 = STRIDEF / 4;                           // stride in float4

  float4 acc = make_float4(0.f, 0.f, 0.f, 0.f);
#pragma unroll
  for (int t = 0; t < 7; ++t) {
    const int p = pos + t - 3;
    if (0 <= p && p < DIM) {                  // SAME zero padding
      const float4 v = in[tid + (long)(t - 3) * S4];
      const float  g = GK[t];
      acc.x += g * v.x; acc.y += g * v.y;
      acc.z += g * v.z; acc.w += g * v.w;
    }
  }
  out[tid] = acc;
}

// ============================================================================
// Launch: x-blur (WMMA) in -> d_out ; y-blur d_out -> d_ws ; z-blur d_ws -> d_out
// d_ws needs out_size * 4 bytes (65.5 MB).
// ============================================================================
extern "C" void kernel_launch(void* const* d_in, const int* in_sizes, int n_in,
                              void* d_out, int out_size, void* d_ws, size_t ws_size,
                              hipStream_t stream) {
  (void)in_sizes; (void)n_in; (void)out_size; (void)ws_size;
  const float* x   = (const float*)d_in[0];
  float*       out = (float*)d_out;
  float*       tmp = (float*)d_ws;

  // 640 slices * 100 tiles = 64000 waves, 4 waves (tiles) per 128-thread block
  gb3d_blurx_wmma<<<SLICES * TILES_PER_SLICE / 4, 128, 0, stream>>>(x, out);

  // 16,384,000 floats / 4 = 4,096,000 float4 = 16000 blocks * 256 (exact)
  const int n4blocks = (4 * DIM * DIM * DIM) / 4 / 256;
  gb3d_blur_axis_f4<DIM><<<n4blocks, 256, 0, stream>>>(
      (const float4*)out, (float4*)tmp);                    // y-blur
  gb3d_blur_axis_f4<DIM * DIM><<<n4blocks, 256, 0, stream>>>(
      (const float4*)tmp, (float4*)out);                    // z-blur
}